// RelayLayer_21285857919544
// MI455X (gfx1250) — compile-verified
//
#include <hip/hip_runtime.h>
#include <math.h>

#if defined(__has_include)
#if __has_include(<hip/amd_detail/amd_gfx1250_TDM.h>)
#define TDM_6ARG 1
#endif
#endif

#if defined(__has_builtin)
#if __has_builtin(__builtin_amdgcn_tensor_load_to_lds)
#define HAVE_TDM 1
#endif
#endif

typedef __attribute__((ext_vector_type(2))) float v2f;
typedef __attribute__((ext_vector_type(4))) float v4f;
typedef __attribute__((ext_vector_type(8))) float v8f;
typedef __attribute__((ext_vector_type(4))) unsigned v4u;
typedef __attribute__((ext_vector_type(4))) int v4i;
typedef __attribute__((ext_vector_type(8))) int v8i;

#define DDIM 4096
#define PNUM 256
#define BROWS 8192
#define NW 8
#define ITER 4

__device__ __forceinline__ v8f wmma_f32_k4(v2f a, v2f b, v8f c) {
  return __builtin_amdgcn_wmma_f32_16x16x4_f32(false, a, false, b, (short)0, c,
                                               false, false);
}

// DPP cross-lane add helpers (no LDS, no waits)
template <int CTRL>
__device__ __forceinline__ float dpp_add(float v) {
  int t = __builtin_amdgcn_update_dpp(0, __float_as_int(v), CTRL, 0xf, 0xf, true);
  return v + __int_as_float(t);
}
// Sum across each 16-lane row via cyclic rotate-add (row_ror:1,2,4,8)
__device__ __forceinline__ float row16_sum(float v) {
  v = dpp_add<0x121>(v);
  v = dpp_add<0x122>(v);
  v = dpp_add<0x124>(v);
  v = dpp_add<0x128>(v);
  return v;
}

#ifdef HAVE_TDM
// Tensor Data Mover: 1-D fp32 block, global -> LDS (D# per ISA 08 sec.8)
__device__ __forceinline__ void tdm_load_f32_1d(void* lds_dst, const float* src,
                                                unsigned nelem) {
  unsigned long long ga = (unsigned long long)(uintptr_t)src;
  unsigned lds = (unsigned)(uintptr_t)lds_dst;  // low 32 bits = LDS offset
  v4u g0 = {0u, 0u, 0u, 0u};
  g0[0] = 1u;                                   // count=1 (valid), user mode
  g0[1] = lds;                                  // lds_addr
  g0[2] = (unsigned)ga;                         // global_addr[31:0]
  g0[3] = (unsigned)((ga >> 32) & 0x01FFFFFFu) | (2u << 30);  // addr[56:32], type=2
  v8i g1 = {0, 0, 0, 0, 0, 0, 0, 0};
  g1[0] = (int)(2u << 16);                      // data_size = 4 bytes
  g1[1] = (int)((nelem & 0xFFFFu) << 16);       // tensor_dim0[15:0]  (bits 63:48)
  g1[2] = (int)((nelem >> 16) & 0xFFFFu);       // tensor_dim0[31:16]
  g1[3] = (int)((nelem & 0xFFFFu) << 16);       // tile_dim0          (bits 127:112)
  g1[5] = (int)nelem;                           // tensor_dim0_stride[31:0]
  v4i gz = {0, 0, 0, 0};
#ifdef TDM_6ARG
  v8i gz8 = {0, 0, 0, 0, 0, 0, 0, 0};
  __builtin_amdgcn_tensor_load_to_lds(g0, g1, gz, gz, gz8, 0);
#else
  __builtin_amdgcn_tensor_load_to_lds(g0, g1, gz, gz, 0);
#endif
}
#endif

// ---------------------------------------------------------------------------
// Precompute slerp anchor directions, transposed: atT[p][k][t*16+a]
// ---------------------------------------------------------------------------
__global__ __launch_bounds__(256) void at_kernel(const float* __restrict__ home,
                                                 const float* __restrict__ anchors,
                                                 float* __restrict__ atT) {
  int idx = blockIdx.x * 256 + threadIdx.x;
  if (idx >= PNUM * 16) return;
  int p = idx >> 4, a = idx & 15;
  const float* hp = home + (size_t)(p * 16 + a) * 16;
  const float* cp = anchors + (size_t)(p * 16 + a) * 16;
  float h[16], c[16];
  float nh = 0.f, nc = 0.f;
#pragma unroll
  for (int k = 0; k < 16; ++k) {
    h[k] = hp[k];
    c[k] = cp[k];
    nh += h[k] * h[k];
    nc += c[k] * c[k];
  }
  float ih = 1.f / fmaxf(sqrtf(nh), 1e-12f);
  float ic = 1.f / fmaxf(sqrtf(nc), 1e-12f);
  float cosv = 0.f;
#pragma unroll
  for (int k = 0; k < 16; ++k) {
    h[k] *= ih;
    c[k] *= ic;
    cosv += h[k] * c[k];
  }
  cosv = fminf(fmaxf(cosv, -1.f + 1e-7f), 1.f - 1e-7f);
  float omega = acosf(cosv);
  float so = fmaxf(sinf(omega), 1e-7f);
#pragma unroll
  for (int t = 0; t < 3; ++t) {
    float tt = (float)t * (1.f / 3.f);
    float s0 = sinf((1.f - tt) * omega) / so;
    float s1 = sinf(tt * omega) / so;
    float w[16];
    float nw = 0.f;
#pragma unroll
    for (int k = 0; k < 16; ++k) {
      w[k] = s0 * h[k] + s1 * c[k];
      nw += w[k] * w[k];
    }
    float iw = 1.f / fmaxf(sqrtf(nw), 1e-12f);
#pragma unroll
    for (int k = 0; k < 16; ++k)
      atT[(size_t)p * 768 + k * 48 + t * 16 + a] = w[k] * iw;
  }
}

// ---------------------------------------------------------------------------
// Per-row LayerNorm stats over D=4096: stats[b] = {mean, rstd}
// ---------------------------------------------------------------------------
__global__ __launch_bounds__(256) void stats_kernel(const float* __restrict__ x,
                                                    float* __restrict__ stats) {
  __shared__ float rs[256], rq[256];
  int b = blockIdx.x;
  int tid = threadIdx.x;
  const float* row = x + (size_t)b * DDIM + tid * 16;
  float s = 0.f, q = 0.f;
#pragma unroll
  for (int j = 0; j < 4; ++j) {
    v4f v = *(const v4f*)(row + j * 4);
#pragma unroll
    for (int e = 0; e < 4; ++e) {
      s += v[e];
      q += v[e] * v[e];
    }
  }
  rs[tid] = s;
  rq[tid] = q;
  __syncthreads();
  for (int off = 128; off > 0; off >>= 1) {
    if (tid < off) {
      rs[tid] += rs[tid + off];
      rq[tid] += rq[tid + off];
    }
    __syncthreads();
  }
  if (tid == 0) {
    float mean = rs[0] * (1.f / DDIM);
    float var = rq[0] * (1.f / DDIM) - mean * mean;
    stats[2 * b] = mean;
    stats[2 * b + 1] = rsqrtf(var + 1e-5f);
  }
}

// ---------------------------------------------------------------------------
// Main fused kernel. Grid (P, 16). Block = 256 (8 waves).
// ---------------------------------------------------------------------------
__global__ __launch_bounds__(256) void relay_main(
    const float* __restrict__ x, const float* __restrict__ atT,
    const float* __restrict__ stats, const float* __restrict__ w1,
    const float* __restrict__ b1, const float* __restrict__ w2,
    const float* __restrict__ b2, const float* __restrict__ png,
    const float* __restrict__ pnb, const float* __restrict__ gates,
    const float* __restrict__ ng, const float* __restrict__ nb,
    float* __restrict__ out) {
  __shared__ float s_at[16 * 48];
  __shared__ float s_w1[48 * 32];
  __shared__ float s_w2[32 * 16];
  __shared__ float s_b1[32];
  __shared__ float s_b2[16];
  __shared__ float s_png[16], s_pnb[16], s_ng[16], s_nb[16];
  __shared__ __align__(16) float s_xn[NW][16][20];
  __shared__ __align__(16) float s_xr[NW][16][20];
  __shared__ float s_tri[NW][16][52];
  __shared__ float s_h[NW][16][36];
  __shared__ float s_inv[NW][16];

  const int tid = threadIdx.x;
  const int p = blockIdx.x;

  // ---- stage per-patch weights into LDS ----
#ifdef HAVE_TDM
  if (tid < 32) {
    tdm_load_f32_1d(s_at, atT + (size_t)p * 768, 768);
    tdm_load_f32_1d(s_w1, w1 + (size_t)p * 1536, 1536);
    tdm_load_f32_1d(s_w2, w2 + (size_t)p * 512, 512);
    __builtin_amdgcn_s_wait_tensorcnt(0);
  }
#else
  for (int i = tid; i < 16 * 48; i += 256) s_at[i] = atT[(size_t)p * 768 + i];
  for (int i = tid; i < 48 * 32; i += 256) s_w1[i] = w1[(size_t)p * 1536 + i];
  for (int i = tid; i < 32 * 16; i += 256) s_w2[i] = w2[(size_t)p * 512 + i];
#endif
  if (tid < 32) s_b1[tid] = b1[(size_t)p * 32 + tid];
  if (tid < 16) {
    s_b2[tid] = b2[(size_t)p * 16 + tid];
    s_png[tid] = png[tid];
    s_pnb[tid] = pnb[tid];
    s_ng[tid] = ng[(size_t)p * 16 + tid];
    s_nb[tid] = nb[(size_t)p * 16 + tid];
  }
  __syncthreads();

  const int wave = tid >> 5;
  const int lane = tid & 31;
  const int half = lane >> 4;
  const int li = lane & 15;
  const float gate = 1.f / (1.f + __expf(-gates[p]));
  const v8f vzero = {0.f, 0.f, 0.f, 0.f, 0.f, 0.f, 0.f, 0.f};

  // ---- hoist all loop-invariant B-operands + per-lane constants ----
  v2f Bat[4][3];
#pragma unroll
  for (int j = 0; j < 4; ++j) {
    const int k0 = 4 * j + 2 * half;
#pragma unroll
    for (int nt = 0; nt < 3; ++nt) {
      Bat[j][nt].x = s_at[k0 * 48 + nt * 16 + li];
      Bat[j][nt].y = s_at[(k0 + 1) * 48 + nt * 16 + li];
    }
  }
  v2f Bw1[12][2];
#pragma unroll
  for (int j = 0; j < 12; ++j) {
    const int k0 = 4 * j + 2 * half;
#pragma unroll
    for (int nt = 0; nt < 2; ++nt) {
      Bw1[j][nt].x = s_w1[k0 * 32 + nt * 16 + li];
      Bw1[j][nt].y = s_w1[(k0 + 1) * 32 + nt * 16 + li];
    }
  }
  v2f Bw2[8];
#pragma unroll
  for (int j = 0; j < 8; ++j) {
    const int k0 = 4 * j + 2 * half;
    Bw2[j].x = s_w2[k0 * 16 + li];
    Bw2[j].y = s_w2[(k0 + 1) * 16 + li];
  }
  const float bb0 = s_b1[li], bb1 = s_b1[16 + li], b2v = s_b2[li];
  const float gn = s_png[li], gb = s_pnb[li];
  const int lrow = lane >> 1;
  const int coff = (lane & 1) * 8;
  float gx[8], bx[8];
#pragma unroll
  for (int e = 0; e < 8; ++e) {
    gx[e] = s_ng[coff + e];
    bx[e] = s_nb[coff + e];
  }

  for (int it = 0; it < ITER; ++it) {
    const int b0 = blockIdx.y * (NW * 16 * ITER) + (it * NW + wave) * 16;

    // ---- load & full-row layernorm of the 16x16 x tile ----
    {
      const float* xp = x + (size_t)(b0 + lrow) * DDIM + p * 16 + coff;
      __builtin_prefetch(xp + (size_t)NW * 16 * DDIM, 0, 1);  // next tile
      v4f u0 = *(const v4f*)xp;
      v4f u1 = *(const v4f*)(xp + 4);
      float mean = stats[2 * (b0 + lrow)];
      float rstd = stats[2 * (b0 + lrow) + 1];
      v4f n0, n1;
      float ss = 0.f;
#pragma unroll
      for (int e = 0; e < 4; ++e) {
        float vn = (u0[e] - mean) * rstd * gx[e] + bx[e];
        n0[e] = vn;
        ss += vn * vn;
      }
#pragma unroll
      for (int e = 0; e < 4; ++e) {
        float vn = (u1[e] - mean) * rstd * gx[4 + e] + bx[4 + e];
        n1[e] = vn;
        ss += vn * vn;
      }
      *(v4f*)&s_xr[wave][lrow][coff] = u0;
      *(v4f*)&s_xr[wave][lrow][coff + 4] = u1;
      *(v4f*)&s_xn[wave][lrow][coff] = n0;
      *(v4f*)&s_xn[wave][lrow][coff + 4] = n1;
      ss = dpp_add<0xB1>(ss);  // quad_perm(1,0,3,2): pair sum
      if ((lane & 1) == 0) s_inv[wave][lrow] = 1.f / fmaxf(sqrtf(ss), 1e-12f);
    }
    __builtin_amdgcn_wave_barrier();

    // ---- GEMM1: tri = 1 - patches_n(16x16) @ atT(16x48) ----
    const float inv = s_inv[wave][li];
    v8f acc0 = vzero, acc1 = vzero, acc2 = vzero;
#pragma unroll
    for (int j = 0; j < 4; ++j) {
      const int k0 = 4 * j + 2 * half;
      v2f a;
      a.x = s_xn[wave][li][k0] * inv;
      a.y = s_xn[wave][li][k0 + 1] * inv;
      acc0 = wmma_f32_k4(a, Bat[j][0], acc0);
      acc1 = wmma_f32_k4(a, Bat[j][1], acc1);
      acc2 = wmma_f32_k4(a, Bat[j][2], acc2);
    }
#pragma unroll
    for (int v = 0; v < 8; ++v) {
      const int m = half * 8 + v;
      s_tri[wave][m][li] = 1.f - acc0[v];
      s_tri[wave][m][16 + li] = 1.f - acc1[v];
      s_tri[wave][m][32 + li] = 1.f - acc2[v];
    }
    __builtin_amdgcn_wave_barrier();

    // ---- GEMM2: hmid = gelu(tri(16x48) @ W1(48x32) + b1) ----
    v8f h0 = vzero, h1 = vzero;
#pragma unroll
    for (int j = 0; j < 12; ++j) {
      const int k0 = 4 * j + 2 * half;
      v2f a;
      a.x = s_tri[wave][li][k0];
      a.y = s_tri[wave][li][k0 + 1];
      h0 = wmma_f32_k4(a, Bw1[j][0], h0);
      h1 = wmma_f32_k4(a, Bw1[j][1], h1);
    }
#pragma unroll
    for (int v = 0; v < 8; ++v) {
      const int m = half * 8 + v;
      float u = h0[v] + bb0;
      s_h[wave][m][li] = 0.5f * u * (1.f + erff(u * 0.70710678f));
      u = h1[v] + bb1;
      s_h[wave][m][16 + li] = 0.5f * u * (1.f + erff(u * 0.70710678f));
    }
    __builtin_amdgcn_wave_barrier();

    // ---- GEMM3: y = hmid(16x32) @ W2(32x16) + b2 ----
    v8f y = vzero;
#pragma unroll
    for (int j = 0; j < 8; ++j) {
      const int k0 = 4 * j + 2 * half;
      v2f a;
      a.x = s_h[wave][li][k0];
      a.y = s_h[wave][li][k0 + 1];
      y = wmma_f32_k4(a, Bw2[j], y);
    }
    float yv[8];
#pragma unroll
    for (int v = 0; v < 8; ++v) yv[v] = y[v] + b2v;

    // ---- layernorm over d=16 via DPP row reductions, blend, store ----
#pragma unroll
    for (int v = 0; v < 8; ++v) {
      float s = row16_sum(yv[v]);
      float mean = s * (1.f / 16.f);
      float c = yv[v] - mean;
      float q = row16_sum(c * c);
      float rstd = rsqrtf(q * (1.f / 16.f) + 1e-5f);
      float pw = c * rstd * gn + gb;
      const int m = half * 8 + v;
      float pv = s_xn[wave][m][li];
      float xr = s_xr[wave][m][li];
      out[(size_t)(b0 + m) * DDIM + p * 16 + li] =
          xr + gate * pw + (1.f - gate) * pv;
    }
    __builtin_amdgcn_wave_barrier();
  }
}

// ---------------------------------------------------------------------------
extern "C" void kernel_launch(void* const* d_in, const int* in_sizes, int n_in,
                              void* d_out, int out_size, void* d_ws,
                              size_t ws_size, hipStream_t stream) {
  const float* x = (const float*)d_in[0];
  const float* home = (const float*)d_in[1];
  const float* anchors = (const float*)d_in[2];
  const float* pw_w1 = (const float*)d_in[3];
  const float* pw_b1 = (const float*)d_in[4];
  const float* pw_w2 = (const float*)d_in[5];
  const float* pw_b2 = (const float*)d_in[6];
  const float* png = (const float*)d_in[7];
  const float* pnb = (const float*)d_in[8];
  const float* gates = (const float*)d_in[9];
  const float* ng = (const float*)d_in[10];
  const float* nb = (const float*)d_in[11];
  float* out = (float*)d_out;

  float* atT = (float*)d_ws;            // 256*16*48 floats
  float* stats = atT + PNUM * 16 * 48;  // 8192*2 floats

  at_kernel<<<16, 256, 0, stream>>>(home, anchors, atT);
  stats_kernel<<<BROWS, 256, 0, stream>>>(x, stats);

  dim3 grid(PNUM, BROWS / (NW * 16 * ITER));
  relay_main<<<grid, 256, 0, stream>>>(x, atT, stats, pw_w1, pw_b1, pw_w2,
                                       pw_b2, png, pnb, gates, ng, nb, out);
}